// MultiStreamSNN_12438225289666
// MI455X (gfx1250) — compile-verified
//
#include <hip/hip_runtime.h>
#include <hip/hip_bf16.h>
#include <math.h>

#define DECAY 0.9f
#define LN_EPS 1e-5f

typedef __attribute__((ext_vector_type(2))) float v2f;
typedef __attribute__((ext_vector_type(8))) float v8f;

__device__ __forceinline__ float lif_sigmoid(float mem) {
    return 1.0f / (1.0f + __expf(-5.0f * (mem - 1.0f)));
}

// ---------------------------------------------------------------------------
// K1: fused encoder GEMM + LIF scan #1.
// grid = 512 blocks (one per chain b*32+a), block = 64 threads (one per E chan).
// Spikes written directly into [b, t, a*64+e] layout (the GEMM's K layout).
// pre[] for the whole 16-step tile is computed before the dependent LIF chain
// so the dot-products and the serial recurrence overlap.
// ---------------------------------------------------------------------------
__global__ void enc_lif_kernel(const float* __restrict__ x,
                               const float* __restrict__ enc_w,
                               const float* __restrict__ enc_b,
                               float* __restrict__ S) {
    const int chain = blockIdx.x;      // b*32 + a
    const int bb = chain >> 5;
    const int aa = chain & 31;
    const int e  = threadIdx.x;        // 0..63

    __shared__ float xs[16][16];       // 16 t-steps x 16 features

    float w[16];
#pragma unroll
    for (int f = 0; f < 16; ++f) w[f] = enc_w[e * 16 + f];
    const float bias = enc_b[e];

    float mem = 0.0f, spk = 0.0f;

    for (int t0 = 0; t0 < 512; t0 += 16) {
        __syncthreads();
        // cooperative load of 16x16 x-tile: x[b, t0+tl, a, f]
        for (int i = e; i < 256; i += 64) {
            const int tl = i >> 4, f = i & 15;
            xs[tl][f] = x[((((size_t)bb * 512) + (t0 + tl)) * 32 + aa) * 16 + f];
        }
        __syncthreads();

        float pre[16];
#pragma unroll
        for (int tl = 0; tl < 16; ++tl) {
            float s = bias;
#pragma unroll
            for (int f = 0; f < 16; ++f) s += xs[tl][f] * w[f];
            pre[tl] = s;
        }
#pragma unroll
        for (int tl = 0; tl < 16; ++tl) {
            mem = DECAY * mem + pre[tl] - spk;
            spk = lif_sigmoid(mem);
            S[((size_t)(bb * 512 + t0 + tl) * 2048) + aa * 64 + e] = spk;
        }
    }
}

// ---------------------------------------------------------------------------
// K2: top-128 of |raw| (1024 elems) by rank counting (stable tie-break on
// index, matching jax.lax.top_k), then c = tanh(raw * mask).
// single block, 1024 threads.
// ---------------------------------------------------------------------------
__global__ void topk_c_kernel(const float* __restrict__ raw,
                              float* __restrict__ c_out,
                              float* __restrict__ c_ws) {
    __shared__ float av[1024];
    const int i = threadIdx.x;
    const float v = raw[i];
    const float a = fabsf(v);
    av[i] = a;
    __syncthreads();
    int cnt = 0;
    for (int j = 0; j < 1024; ++j) {
        const float aj = av[j];
        cnt += (aj > a) || (aj == a && j < i);
    }
    const float cv = (cnt < 128) ? tanhf(v) : 0.0f;
    c_out[i] = cv;
    c_ws[i]  = cv;
}

// ---------------------------------------------------------------------------
// K3: fold c into fusion weights: Weff[f, j*64+d] = sum_i fus_w[f,i*64+d]*c[i,j]
// 524288 outputs, 32 MACs each. grid = 2048 x 256.
// ---------------------------------------------------------------------------
__global__ void weff_kernel(const float* __restrict__ fus_w,
                            const float* __restrict__ c,
                            float* __restrict__ Weff) {
    __shared__ float cs[1024];
    for (int i = threadIdx.x; i < 1024; i += blockDim.x) cs[i] = c[i];
    __syncthreads();
    const int k = blockIdx.x * blockDim.x + threadIdx.x;
    const int f = k >> 11;
    const int r = k & 2047;
    const int j = r >> 6;
    const int d = r & 63;
    float s = 0.0f;
#pragma unroll
    for (int i = 0; i < 32; ++i)
        s += fus_w[f * 2048 + i * 64 + d] * cs[i * 32 + j];
    Weff[k] = s;
}

// ---------------------------------------------------------------------------
// K4: the big GEMM on WMMA f32 16x16x4 with 4x2 register tiling:
//   P[m, n] = sum_k S[m, k] * Weff[n, k] + fus_b[n],  M=8192, N=256, K=2048.
// Each wave owns a 64x32 output macro-tile (4 M-tiles x 2 N-tiles, 8 v8f
// accumulators). Per K=4 step: 6 float2 fragment loads feed 8 WMMAs
// (1.5 floats/lane/wmma vs 4.0 for the unblocked version -> 2.7x less L2
// traffic). 1024 waves total keeps latency hiding alive.
// A-frag (16x4 f32): lanes 0-15 -> {K=k0,k0+1}, lanes 16-31 -> {K=k0+2,k0+3}.
// B-frag (4x16 f32): same K split across lane halves, N = lane&15.
// C/D layout: VGPR r -> M = r + 8*(lane>=16), N = lane&15 (ISA 7.12.2).
// ---------------------------------------------------------------------------
__global__ void fus_gemm_wmma(const float* __restrict__ S,
                              const float* __restrict__ Weff,
                              const float* __restrict__ fus_b,
                              float* __restrict__ P) {
    const int gwave = (int)((blockIdx.x * blockDim.x + threadIdx.x) >> 5); // 0..1023
    const int lane  = threadIdx.x & 31;
    const int tm = gwave >> 3;          // 128 macro M-tiles (64 rows each)
    const int tn = gwave & 7;           // 8 macro N-tiles (32 cols each)
    const int m0 = tm << 6;
    const int n0 = tn << 5;
    const int l  = lane & 15;
    const int hi = lane >> 4;

    const float* abase = S    + (size_t)(m0 + l) * 2048 + 2 * hi;
    const float* bbase = Weff + (size_t)(n0 + l) * 2048 + 2 * hi;

    v8f acc[4][2] = {};

#pragma unroll 2
    for (int k0 = 0; k0 < 2048; k0 += 4) {
        v2f af[4], bf[2];
#pragma unroll
        for (int i = 0; i < 4; ++i)
            af[i] = *(const v2f*)(abase + (size_t)i * 16 * 2048 + k0);
#pragma unroll
        for (int j = 0; j < 2; ++j)
            bf[j] = *(const v2f*)(bbase + (size_t)j * 16 * 2048 + k0);
#pragma unroll
        for (int i = 0; i < 4; ++i)
#pragma unroll
            for (int j = 0; j < 2; ++j)
                acc[i][j] = __builtin_amdgcn_wmma_f32_16x16x4_f32(
                    /*neg_a=*/false, af[i], /*neg_b=*/false, bf[j],
                    /*c_mod=*/(short)0, acc[i][j],
                    /*reuse_a=*/false, /*reuse_b=*/false);
    }

#pragma unroll
    for (int i = 0; i < 4; ++i) {
#pragma unroll
        for (int j = 0; j < 2; ++j) {
            const int col = n0 + j * 16 + l;
            const float bias = fus_b[col];
            float* out = P + (size_t)(m0 + i * 16 + 8 * hi) * 256 + col;
#pragma unroll
            for (int r = 0; r < 8; ++r) out[(size_t)r * 256] = acc[i][j][r] + bias;
        }
    }
}

// ---------------------------------------------------------------------------
// K5: LIF scan #2 + LayerNorm(last) + head.
// grid = 16 blocks (batch), block = 256 threads (channel).
// Loads are batched 8 t-steps at a time so they issue as one clause and the
// (serial) recurrence only pays L2 latency once per 8 steps.
// ---------------------------------------------------------------------------
__global__ void fus_scan_head(const float* __restrict__ P,
                              const float* __restrict__ ln_g,
                              const float* __restrict__ ln_b,
                              const float* __restrict__ head_w,
                              const float* __restrict__ head_b,
                              float* __restrict__ out_logits,
                              float* __restrict__ out_fused) {
    const int b = blockIdx.x;
    const int f = threadIdx.x;

    float mem = 0.0f, spk = 0.0f;
    const float* p = P         + (size_t)b * 512 * 256 + f;
    float*       o = out_fused + (size_t)b * 512 * 256 + f;

    for (int t0 = 0; t0 < 512; t0 += 8) {
        float pre[8];
#pragma unroll
        for (int i = 0; i < 8; ++i) pre[i] = p[(size_t)(t0 + i) * 256];
#pragma unroll
        for (int i = 0; i < 8; ++i) {
            mem = DECAY * mem + pre[i] - spk;
            spk = lif_sigmoid(mem);
            o[(size_t)(t0 + i) * 256] = spk;
        }
    }

    // LayerNorm over 256 channels of the last spike (population variance)
    __shared__ float red[256];
    __shared__ float red2[256];
    red[f]  = spk;
    red2[f] = spk * spk;
    __syncthreads();
    for (int s = 128; s > 0; s >>= 1) {
        if (f < s) { red[f] += red[f + s]; red2[f] += red2[f + s]; }
        __syncthreads();
    }
    const float mu  = red[0]  * (1.0f / 256.0f);
    const float var = red2[0] * (1.0f / 256.0f) - mu * mu;
    const float inv = rsqrtf(var + LN_EPS);

    __shared__ float h[256];
    h[f] = (spk - mu) * inv * ln_g[f] + ln_b[f];
    __syncthreads();

    if (f < 8) {
        float s = head_b[f];
        for (int k = 0; k < 256; ++k) s += h[k] * head_w[f * 256 + k];
        out_logits[b * 8 + f] = s;
    }
}

// ---------------------------------------------------------------------------
extern "C" void kernel_launch(void* const* d_in, const int* in_sizes, int n_in,
                              void* d_out, int out_size, void* d_ws, size_t ws_size,
                              hipStream_t stream) {
    const float* x      = (const float*)d_in[0];
    const float* enc_w  = (const float*)d_in[1];
    const float* enc_b  = (const float*)d_in[2];
    const float* raw    = (const float*)d_in[3];
    const float* fus_w  = (const float*)d_in[4];
    const float* fus_b  = (const float*)d_in[5];
    const float* ln_g   = (const float*)d_in[6];
    const float* ln_b   = (const float*)d_in[7];
    const float* head_w = (const float*)d_in[8];
    const float* head_b = (const float*)d_in[9];

    float* ws   = (float*)d_ws;
    float* S    = ws;                              // 16*512*2048 = 16777216 floats
    float* Weff = ws + 16777216;                   // 256*2048    =   524288 floats
    float* c_ws = ws + 16777216 + 524288;          // 1024 floats
    float* P    = ws + 16777216 + 524288 + 1024;   // 16*512*256  =  2097152 floats

    float* out        = (float*)d_out;
    float* out_logits = out;          // 16*8   = 128
    float* out_c      = out + 128;    // 32*32  = 1024
    float* out_fused  = out + 1152;   // 16*512*256

    // K1: encoder GEMM + LIF scan 1
    enc_lif_kernel<<<512, 64, 0, stream>>>(x, enc_w, enc_b, S);
    // K2: top-k mask + c = tanh(raw*mask)
    topk_c_kernel<<<1, 1024, 0, stream>>>(raw, out_c, c_ws);
    // K3: fold c into fusion weights
    weff_kernel<<<2048, 256, 0, stream>>>(fus_w, c_ws, Weff);
    // K4: big WMMA GEMM -> fus_pre (1024 waves, 4x2 register-tiled)
    fus_gemm_wmma<<<256, 128, 0, stream>>>(S, Weff, fus_b, P);
    // K5: LIF scan 2 + LayerNorm + head
    fus_scan_head<<<16, 256, 0, stream>>>(P, ln_g, ln_b, head_w, head_b,
                                          out_logits, out_fused);
}